// GCN_29025388986650
// MI455X (gfx1250) — compile-verified
//
#include <hip/hip_runtime.h>
#include <math.h>

typedef __attribute__((ext_vector_type(2))) float v2f;
typedef __attribute__((ext_vector_type(8))) float v8f;

#define FDIM 128   // input features
#define HDIM 32    // hidden features

// ---------------------------------------------------------------------------
// Degree / normalization
// ---------------------------------------------------------------------------
__global__ void zero_f32(float* __restrict__ p, int n) {
    int i = blockIdx.x * blockDim.x + threadIdx.x;
    if (i < n) p[i] = 0.0f;
}

__global__ void count_deg(const int* __restrict__ ei, float* __restrict__ deg, int E) {
    int e = blockIdx.x * blockDim.x + threadIdx.x;
    if (e < E) atomicAdd(&deg[ei[E + e]], 1.0f);   // dst row of edge_index
}

// deg -> dinv in place; every node has exactly one self loop, hence +1 (deg>0 always)
__global__ void make_dinv(float* __restrict__ deg, int n) {
    int i = blockIdx.x * blockDim.x + threadIdx.x;
    if (i < n) deg[i] = rsqrtf(deg[i] + 1.0f);
}

// ---------------------------------------------------------------------------
// Layer-1 GEMM: h1[N,32] = x[N,128] @ W1[128,32]  -- full-precision f32 WMMA.
// One wave32 per 16x16 output tile. N % 16 == 0 (100000 = 6250*16).
// A (16x4 f32): lanes 0-15 hold K = {0,1}, lanes 16-31 hold K = {2,3}.
// B (4x16 f32): VGPR0 = rows K0/K2, VGPR1 = rows K1/K3, striped across lanes.
// ---------------------------------------------------------------------------
__global__ __launch_bounds__(32) void gemm1_wmma(const float* __restrict__ x,
                                                 const float* __restrict__ W1,
                                                 float* __restrict__ h1) {
    const int rowBase = blockIdx.x * 16;          // M tile
    const int nBase   = blockIdx.y * 16;          // N tile (0 or 16)
    const int lane = threadIdx.x;
    const int half = lane >> 4;                   // 0: lanes 0-15, 1: lanes 16-31
    const int l16  = lane & 15;

    const float* xrow = x  + (size_t)(rowBase + l16) * FDIM + 2 * half; // K offset 0 or 2
    const float* wcol = W1 + (size_t)(2 * half) * HDIM + nBase + l16;

    v8f acc = {};
#pragma unroll
    for (int k = 0; k < FDIM; k += 4) {
        v2f a;
        a.x = xrow[k];                 // A: x[row, k + 2*half + 0]
        a.y = xrow[k + 1];             //    x[row, k + 2*half + 1]
        v2f b;
        b.x = wcol[(size_t)k * HDIM];            // B: W1[k + 2*half + 0, n]
        b.y = wcol[(size_t)(k + 1) * HDIM];      //    W1[k + 2*half + 1, n]
        acc = __builtin_amdgcn_wmma_f32_16x16x4_f32(
            /*neg_a=*/false, a, /*neg_b=*/false, b,
            /*c_mod=*/(short)0, acc, /*reuse_a=*/false, /*reuse_b=*/false);
    }

    // D layout: VGPR j holds M = j + 8*half, N = l16
    float* outp = h1 + (size_t)rowBase * HDIM + nBase + l16;
#pragma unroll
    for (int j = 0; j < 8; ++j)
        outp[(size_t)(j + 8 * half) * HDIM] = acc[j];
}

// ---------------------------------------------------------------------------
// agg1 init with self-loop term: agg1[i,f] = h1[i,f] * dinv[i]^2
// (also fully initializes the 0xAA-poisoned workspace region)
// ---------------------------------------------------------------------------
__global__ void init_agg1(const float* __restrict__ h1, const float* __restrict__ dinv,
                          float* __restrict__ agg1, int n) {
    int gid = blockIdx.x * blockDim.x + threadIdx.x;   // n * 32 threads
    int i = gid >> 5;
    if (i < n) {
        float d = dinv[i];
        agg1[gid] = h1[gid] * (d * d);
    }
}

// ---------------------------------------------------------------------------
// Layer-1 edge scatter: one wave32 per edge, one lane per feature.
// Lanes of a wave read h1[src*32 .. +31] (one coalesced 128B line, L2-resident)
// and do 32 coalesced global_atomic_add_f32 into agg1[dst*32 ..].
// ---------------------------------------------------------------------------
__global__ void scatter1(const int* __restrict__ ei, const float* __restrict__ dinv,
                         const float* __restrict__ h1, float* __restrict__ agg1, int E) {
    long long gid = (long long)blockIdx.x * blockDim.x + threadIdx.x;
    int e = (int)(gid >> 5);
    int f = (int)(gid & 31);
    if (e < E) {
        int src = ei[e];
        int dst = ei[E + e];
        float nrm = dinv[src] * dinv[dst];
        atomicAdd(&agg1[(size_t)dst * HDIM + f], h1[(size_t)src * HDIM + f] * nrm);
    }
}

// ---------------------------------------------------------------------------
// Fused: z = relu(agg1 + b1); h2 = z @ W2[32,2]; agg2 = h2 * dinv^2 (self loop)
// ---------------------------------------------------------------------------
__global__ void layer2_fused(const float* __restrict__ agg1, const float* __restrict__ b1,
                             const float* __restrict__ W2, const float* __restrict__ dinv,
                             float* __restrict__ h2, float* __restrict__ agg2, int n) {
    int i = blockIdx.x * blockDim.x + threadIdx.x;
    if (i >= n) return;
    const float* row = agg1 + (size_t)i * HDIM;
    float acc0 = 0.0f, acc1 = 0.0f;
#pragma unroll
    for (int k = 0; k < HDIM; ++k) {
        float z = row[k] + b1[k];
        z = z > 0.0f ? z : 0.0f;               // ReLU
        acc0 += z * W2[k * 2 + 0];
        acc1 += z * W2[k * 2 + 1];
    }
    float d = dinv[i];
    float d2 = d * d;
    h2[(size_t)i * 2 + 0]   = acc0;
    h2[(size_t)i * 2 + 1]   = acc1;
    agg2[(size_t)i * 2 + 0] = acc0 * d2;
    agg2[(size_t)i * 2 + 1] = acc1 * d2;
}

// ---------------------------------------------------------------------------
// Layer-2 edge scatter: one thread per edge, 2 features.
// ---------------------------------------------------------------------------
__global__ void scatter2(const int* __restrict__ ei, const float* __restrict__ dinv,
                         const float* __restrict__ h2, float* __restrict__ agg2, int E) {
    int e = blockIdx.x * blockDim.x + threadIdx.x;
    if (e < E) {
        int src = ei[e];
        int dst = ei[E + e];
        float nrm = dinv[src] * dinv[dst];
        atomicAdd(&agg2[(size_t)dst * 2 + 0], h2[(size_t)src * 2 + 0] * nrm);
        atomicAdd(&agg2[(size_t)dst * 2 + 1], h2[(size_t)src * 2 + 1] * nrm);
    }
}

// ---------------------------------------------------------------------------
// logits = agg2 + b2 ; out = log_softmax(logits, axis=1)  (C = 2)
// ---------------------------------------------------------------------------
__global__ void logsoftmax2(const float* __restrict__ agg2, const float* __restrict__ b2,
                            float* __restrict__ out, int n) {
    int i = blockIdx.x * blockDim.x + threadIdx.x;
    if (i >= n) return;
    float l0 = agg2[(size_t)i * 2 + 0] + b2[0];
    float l1 = agg2[(size_t)i * 2 + 1] + b2[1];
    float m  = l0 > l1 ? l0 : l1;
    float lse = m + logf(__expf(l0 - m) + __expf(l1 - m));
    out[(size_t)i * 2 + 0] = l0 - lse;
    out[(size_t)i * 2 + 1] = l1 - lse;
}

// ---------------------------------------------------------------------------
extern "C" void kernel_launch(void* const* d_in, const int* in_sizes, int n_in,
                              void* d_out, int out_size, void* d_ws, size_t ws_size,
                              hipStream_t stream) {
    const float* x  = (const float*)d_in[0];   // [N,128]
    const int*   ei = (const int*)  d_in[1];   // [2,E] row-major: src row then dst row
    const float* W1 = (const float*)d_in[2];   // [128,32]
    const float* b1 = (const float*)d_in[3];   // [32]
    const float* W2 = (const float*)d_in[4];   // [32,2]
    const float* b2 = (const float*)d_in[5];   // [2]
    float* out = (float*)d_out;                // [N,2] log-softmax

    const int N = in_sizes[0] / FDIM;          // 100000 (divisible by 16)
    const int E = in_sizes[1] / 2;             // 3200000

    // workspace layout (floats): dinv[N] | h1[N*32] | agg1[N*32] | h2[N*2] | agg2[N*2]
    float* ws   = (float*)d_ws;
    float* dinv = ws;
    float* h1   = dinv + N;
    float* agg1 = h1   + (size_t)N * HDIM;
    float* h2   = agg1 + (size_t)N * HDIM;
    float* agg2 = h2   + (size_t)N * 2;

    const int TB = 256;

    // 1) degree (includes +1 self loop inside make_dinv) -> dinv in place
    zero_f32 <<<(N + TB - 1) / TB, TB, 0, stream>>>(dinv, N);
    count_deg<<<(E + TB - 1) / TB, TB, 0, stream>>>(ei, dinv, E);
    make_dinv<<<(N + TB - 1) / TB, TB, 0, stream>>>(dinv, N);

    // 2) h1 = x @ W1 via f32 WMMA (6250 x 2 tiles, one wave per tile)
    gemm1_wmma<<<dim3(N / 16, HDIM / 16), 32, 0, stream>>>(x, W1, h1);

    // 3) agg1 = h1 * dinv^2 (self loops), then edge scatter
    {
        long long t = (long long)N * HDIM;
        init_agg1<<<(unsigned)((t + TB - 1) / TB), TB, 0, stream>>>(h1, dinv, agg1, N);
    }
    {
        long long t = (long long)E * HDIM;
        scatter1<<<(unsigned)((t + TB - 1) / TB), TB, 0, stream>>>(ei, dinv, h1, agg1, E);
    }

    // 4) relu+bias, h2 = z @ W2, agg2 self-loop init (fused)
    layer2_fused<<<(N + TB - 1) / TB, TB, 0, stream>>>(agg1, b1, W2, dinv, h2, agg2, N);

    // 5) layer-2 edge scatter
    scatter2<<<(E + TB - 1) / TB, TB, 0, stream>>>(ei, dinv, h2, agg2, E);

    // 6) bias + log_softmax
    logsoftmax2<<<(N + TB - 1) / TB, TB, 0, stream>>>(agg2, b2, out, N);
}